// NaiveVoxelPooling_27504970564292
// MI455X (gfx1250) — compile-verified
//
#include <hip/hip_runtime.h>

// Problem constants (match reference setup_inputs)
#define NVX 128
#define NVY 128
#define NC  80
#define B_  2
#define N_  6
#define D_  112
#define H_  16
#define W_  44
#define HW  (H_ * W_)            // 704
#define RAYS (B_ * N_ * H_ * W_) // 8448
#define RAYS_PER_BLOCK 8         // 8 waves of 32 -> 256 threads

typedef __attribute__((ext_vector_type(2))) float v2f;
typedef __attribute__((ext_vector_type(8))) float v8f;

__global__ __launch_bounds__(256) void vp_zero(float4* __restrict__ out, int n4) {
  int i = blockIdx.x * blockDim.x + threadIdx.x;
  const int stride = gridDim.x * blockDim.x;
  for (; i < n4; i += stride) out[i] = make_float4(0.f, 0.f, 0.f, 0.f);
}

__global__ __launch_bounds__(256) void vp_pool(const int*   __restrict__ geom,
                                               const float* __restrict__ depth,
                                               const float* __restrict__ ctx,
                                               float*       __restrict__ out) {
  const int lane = threadIdx.x & 31;
  const int wave = threadIdx.x >> 5;
  const int ray  = blockIdx.x * RAYS_PER_BLOCK + wave;  // grid covers exactly RAYS

  const bool lo = lane < 16;
  const int w   = ray % W_;
  const int t   = ray / W_;
  const int h   = t % H_;
  const int bn  = t / H_;
  const int b   = bn / N_;

  // point index p(d) = bn*D*HW + d*HW + h*W + w
  const int baseP   = bn * (D_ * HW) + h * W_ + w;
  // ctx index (b,n,c,h,w) = bn*C*HW + c*HW + h*W + w
  const int ctxBase = bn * (NC * HW) + h * W_ + w;
  const int voxBase = b * (NVX * NVY);

  // Load all 80 context channels for this ray into 5 WMMA B-operands.
  // B(4x16): VGPR0 lanes 0-15 = row K=0 (the only nonzero row).
  v2f Bv[5];
#pragma unroll
  for (int ct = 0; ct < 5; ++ct) {
    float cv = 0.f;
    if (lo) cv = ctx[ctxBase + (ct * 16 + lane) * HW];
    v2f bb; bb.x = cv; bb.y = 0.f;
    Bv[ct] = bb;
  }

  const int rowoff = (lane >> 4) << 3;  // 0 for lanes 0-15, 8 for 16-31
  const int lanec  = lane & 15;         // channel within a 16-wide tile

  for (int dt = 0; dt < 7; ++dt) {
    const int d0 = dt * 16;

    // Per-lane prefetch: lanes 0-15 pull the next depth tile's strided
    // cachelines; lanes 16-31 reach one tile further (clamped).
    if (dt < 6) {
      int pd = d0 + 16 + lane;
      pd = pd < D_ ? pd : (D_ - 1);
      const int pp = baseP + pd * HW;
      __builtin_prefetch(&geom[3 * pp], 0, 1);
      __builtin_prefetch(&depth[pp], 0, 1);
    }

    // A(16x4): VGPR0 lanes 0-15 = column K=0 = 16 depth bins.
    // Reference semantics: invalid points contribute 0.0 at the CLAMPED
    // voxel index (feats zeroed, idx clipped) -> zero dv, clamp coords,
    // scatter unconditionally. No per-atomic branching needed.
    float dv  = 0.f;
    int  voff = 0;  // element offset of voxel row = vox * NC
    if (lo) {
      const int p  = baseP + (d0 + lane) * HW;
      dv = depth[p];
      const int gx = geom[3 * p + 0];
      const int gy = geom[3 * p + 1];
      if (!((unsigned)gx < (unsigned)NVX && (unsigned)gy < (unsigned)NVY))
        dv = 0.f;
      const int cx = gx < 0 ? 0 : (gx > NVX - 1 ? NVX - 1 : gx);
      const int cy = gy < 0 ? 0 : (gy > NVY - 1 ? NVY - 1 : gy);
      voff = (voxBase + cy * NVX + cx) * NC;
    }
    v2f Av; Av.x = dv; Av.y = 0.f;

    // Broadcast the pre-multiplied 32-bit element offsets for rows r / r+8.
    int offr[8];
#pragma unroll
    for (int r = 0; r < 8; ++r)
      offr[r] = __shfl(voff, r + rowoff, 32);

    // All 5 exact f32 outer-product tiles first:
    // D[M][N] = depth[d0+M] * ctx[ct*16+N]
    v8f acc[5];
#pragma unroll
    for (int ct = 0; ct < 5; ++ct) {
      v8f z = {0.f, 0.f, 0.f, 0.f, 0.f, 0.f, 0.f, 0.f};
      acc[ct] = __builtin_amdgcn_wmma_f32_16x16x4_f32(
          /*neg_a=*/false, Av, /*neg_b=*/false, Bv[ct],
          /*c_mod=*/(short)0, z, /*reuse_a=*/false, /*reuse_b=*/false);
    }

    // Scatter: r outer so the per-row base address is computed once;
    // ct*16 folds into the atomic's immediate offset.
#pragma unroll
    for (int r = 0; r < 8; ++r) {
      float* p = out + (unsigned)(offr[r] + lanec);
#pragma unroll
      for (int ct = 0; ct < 5; ++ct) {
        __hip_atomic_fetch_add(p + ct * 16, acc[ct][r], __ATOMIC_RELAXED,
                               __HIP_MEMORY_SCOPE_AGENT);
      }
    }
  }
}

extern "C" void kernel_launch(void* const* d_in, const int* in_sizes, int n_in,
                              void* d_out, int out_size, void* d_ws, size_t ws_size,
                              hipStream_t stream) {
  (void)in_sizes; (void)n_in; (void)d_ws; (void)ws_size;
  const int*   geom  = (const int*)d_in[0];
  const float* depth = (const float*)d_in[1];
  const float* ctx   = (const float*)d_in[2];
  float*       out   = (float*)d_out;

  // Harness poisons d_out: zero the accumulator first (same stream => ordered).
  const int n4 = out_size / 4;   // 2*128*128*80 / 4 = 655,360 float4s
  vp_zero<<<512, 256, 0, stream>>>((float4*)out, n4);

  vp_pool<<<RAYS / RAYS_PER_BLOCK, 256, 0, stream>>>(geom, depth, ctx, out);
}